// GraphPLVEncoder_57260503990388
// MI455X (gfx1250) — compile-verified
//
#include <hip/hip_runtime.h>
#include <hip/hip_bf16.h>
#include <math.h>

typedef __attribute__((ext_vector_type(2))) float v2f;
typedef __attribute__((ext_vector_type(8))) float v8f;

#define BN_EPS 1e-5f

// =====================================================================
// Kernel 0: one-shot transpose of conv2 weights:
//   w2 [c2][c][k] (64x32x9) -> w2T [K=c*9+k][c2]  (288x64)
// =====================================================================
__global__ void transpose_w2_kernel(const float* __restrict__ w2, float* __restrict__ w2T)
{
    int i = blockIdx.x * 256 + threadIdx.x;
    if (i >= 288 * 64) return;
    int K = i / 64, c2 = i % 64;
    int c = K / 9, k = K % 9;
    w2T[i] = w2[(c2 * 32 + c) * 9 + k];
}

// =====================================================================
// Kernel 1: temporal encoder, fully WMMA (f32 16x16x4).
// 4 sequences / block, 256 threads = 8 wave32.
// conv1 as GEMM 400x32 (K=28) with in-register ReLU+maxpool4 epilogue,
// conv2 as GEMM 64x64 (K=288) via im2col, then pool2+mean epilogue.
// The conv2 B-matrix (72KB) is staged with async global->LDS copies
// overlapping im2col1 + the conv1 WMMA phase (ASYNCcnt).
// =====================================================================
__global__ void encoder_kernel(const float* __restrict__ eeg,
                               const float* __restrict__ w1, const float* __restrict__ b1,
                               const float* __restrict__ w2T, const float* __restrict__ b2,
                               float* __restrict__ X, int nSeqTotal)
{
    extern __shared__ float sm[];
    float* s_b1  = sm;                 // 32
    float* s_b2  = s_b1 + 32;          // 64
    float* s_B1  = s_b2 + 64;          // 28 x 40 (stride 40)          = 1120
    float* s_p   = s_B1 + 1120;        // 4 x 32 x 25                  = 3200
    float* s_B2  = s_p + 3200;         // 288 x 64 (compact)           = 18432
    float* s_x   = s_B2 + 18432;       // 4 x 400                      = 1600
    float* s_U   = s_x + 1600;         // union scratch                = 23296
    float* sIm1  = s_U;                // 400 x 28
    float* sIm2  = s_U;                // 64 x 292 (stride 292) = 18688
    float* sOut2 = s_U + 18688;        // 64 x 72 (stride 72)   = 4608
    // total = 47744 floats = 190976 B  (CDNA5: up to 320KB per WG)

    const int tid  = threadIdx.x;
    const int seq0 = blockIdx.x * 4;

    __builtin_prefetch(eeg + (size_t)seq0 * 400, 0, 1);

    // ---- async stage of conv2 B-matrix: 18432 floats, 16B per lane-op ----
    {
        unsigned ldsBase = (unsigned)(uintptr_t)s_B2;   // wave-relative LDS byte offset
        #pragma unroll
        for (int it = 0; it < 18; ++it) {
            unsigned byteOff = (unsigned)(it * 256 + tid) * 16u;
            unsigned ldsAddr = ldsBase + byteOff;
            asm volatile("global_load_async_to_lds_b128 %0, %1, %2"
                         :: "v"(ldsAddr), "v"(byteOff), "s"(w2T)
                         : "memory");
        }
    }

    // ---- stage remaining weights / inputs (synchronous) ----
    if (tid < 32) s_b1[tid] = b1[tid];
    if (tid < 64) s_b2[tid] = b2[tid];
    for (int i = tid; i < 28 * 40; i += 256) {
        int k = i / 40, c = i % 40;
        s_B1[i] = (k < 25 && c < 32) ? w1[c * 25 + k] : 0.f;
    }
    for (int i = tid; i < 1600; i += 256) {
        int s = i / 400, t = i % 400;
        int g = seq0 + s;
        s_x[i] = (g < nSeqTotal) ? eeg[(size_t)g * 400 + t] : 0.f;
    }
    __syncthreads();

    // ---- im2col for conv1: A1[m=seq*100+t][kk] = x[seq][4t-12+kk] ----
    for (int i = tid; i < 400 * 28; i += 256) {
        int m = i / 28, kk = i % 28;
        int seq = m / 100, t = m % 100;
        int j = 4 * t - 12 + kk;
        sIm1[i] = (kk < 25 && (unsigned)j < 400u) ? s_x[seq * 400 + j] : 0.f;
    }
    __syncthreads();

    const int w    = tid >> 5;      // wave 0..7
    const int lane = tid & 31;
    const int l16  = lane & 15;
    const int h    = lane >> 4;

    // ---- conv1 GEMM: (400x28) @ (28x32), epilogue bias+ReLU+maxpool4 ----
    for (int tile = w; tile < 50; tile += 8) {       // wave-uniform trip counts
        int mt = tile >> 1, nt = tile & 1;
        v8f acc = {};
        const float* A = sIm1 + (16 * mt + l16) * 28;
        const float* B = s_B1 + 16 * nt + l16;
        #pragma unroll
        for (int kk = 0; kk < 28; kk += 4) {
            int k0 = kk + 2 * h;
            v2f bf; bf.x = B[k0 * 40];  bf.y = B[(k0 + 1) * 40];
            v2f af; af.x = A[k0];       af.y = A[k0 + 1];
            acc = __builtin_amdgcn_wmma_f32_16x16x4_f32(false, af, false, bf, (short)0, acc, false, false);
        }
        int c = 16 * nt + l16;
        float bias = s_b1[c];
        // C layout: row M = 16*mt + v + 8*h; pool groups of 4 rows are lane-local
        float m0 = fmaxf(fmaxf(acc[0], acc[1]), fmaxf(acc[2], acc[3]));
        float m1 = fmaxf(fmaxf(acc[4], acc[5]), fmaxf(acc[6], acc[7]));
        float p0 = fmaxf(m0 + bias, 0.f);
        float p1 = fmaxf(m1 + bias, 0.f);
        int r0 = 4 * mt + 2 * h;                     // pooled row in [0,100)
        int r1 = r0 + 1;
        s_p[(r0 / 25) * 800 + c * 25 + (r0 % 25)] = p0;
        s_p[(r1 / 25) * 800 + c * 25 + (r1 % 25)] = p1;
    }
    __syncthreads();

    // ---- im2col for conv2: A2[m=seq*13+t2][K=c*9+k] = p[seq][c][2t2-4+k]
    for (int i = tid; i < 64 * 292; i += 256) {
        int m = i / 292, K = i % 292;
        float v = 0.f;
        if (m < 52 && K < 288) {
            int seq = m / 13, t2 = m % 13;
            int c = K / 9, k = K % 9;
            int j = 2 * t2 - 4 + k;
            if ((unsigned)j < 25u) v = s_p[seq * 800 + c * 25 + j];
        }
        sIm2[i] = v;
    }
    // make this wave's async B2 transfers LDS-visible, then barrier for all waves
    asm volatile("s_wait_asynccnt 0" ::: "memory");
    __syncthreads();

    // ---- conv2 GEMM: (64x288) @ (288x64) -> sOut2 ----
    {
        const int nt  = w & 3;
        const int mtb = (w >> 2) << 1;               // 0 or 2
        v8f acc0 = {}; v8f acc1 = {};
        const float* A0 = sIm2 + (16 * mtb + l16) * 292;
        const float* A1 = sIm2 + (16 * (mtb + 1) + l16) * 292;
        const float* B  = s_B2 + 16 * nt + l16;
        #pragma unroll 4
        for (int kk = 0; kk < 288; kk += 4) {
            int k0 = kk + 2 * h;
            v2f bf; bf.x = B[k0 * 64];  bf.y = B[(k0 + 1) * 64];
            v2f a0; a0.x = A0[k0];      a0.y = A0[k0 + 1];
            v2f a1; a1.x = A1[k0];      a1.y = A1[k0 + 1];
            acc0 = __builtin_amdgcn_wmma_f32_16x16x4_f32(false, a0, false, bf, (short)0, acc0, false, false);
            acc1 = __builtin_amdgcn_wmma_f32_16x16x4_f32(false, a1, false, bf, (short)0, acc1, false, false);
        }
        int col = 16 * nt + l16;
        #pragma unroll
        for (int v = 0; v < 8; ++v) {
            sOut2[(16 * mtb + v + 8 * h) * 72 + col]       = acc0[v];
            sOut2[(16 * (mtb + 1) + v + 8 * h) * 72 + col] = acc1[v];
        }
    }
    __syncthreads();

    // ---- epilogue: bias+ReLU+maxpool2+mean(6) ----
    {
        int s = tid / 64, c2 = tid % 64;
        int g = seq0 + s;
        float bias = s_b2[c2];
        float o[13];
        #pragma unroll
        for (int t = 0; t < 13; ++t)
            o[t] = fmaxf(sOut2[(s * 13 + t) * 72 + c2] + bias, 0.f);
        float sum = 0.f;
        #pragma unroll
        for (int t = 0; t < 6; ++t) sum += fmaxf(o[2 * t], o[2 * t + 1]);
        if (g < nSeqTotal) X[(size_t)g * 64 + c2] = sum * (1.f / 6.f);
    }
}

// =====================================================================
// Kernel 2: graph conv GEMM (pre-BN): G = (softmax(q)·A @ X) @ W
// One block per (b,s); 128 threads = 4 wave32; f32 WMMA 16x16x4.
// LDS tiles padded to stride 72 to decouple the two lane-halves' banks.
// =====================================================================
__global__ void graph_gemm_kernel(const float* __restrict__ Xin,
                                  const float* __restrict__ plv,
                                  const float* __restrict__ q,
                                  const float* __restrict__ W,
                                  float* __restrict__ Gout)
{
    __shared__ float sX[20][72];   // X rows 0..18, row 19 zero (K padded to 20)
    __shared__ float sA[32][20];   // Abar padded to 32x20
    __shared__ float sW[64][72];
    __shared__ float sAX[32][72];

    const int tid = threadIdx.x;
    const int bs  = blockIdx.x;         // b*15+s
    const int b   = bs / 15, s = bs % 15;

    __builtin_prefetch(Xin + (size_t)bs * 19 * 64, 0, 1);

    float q0 = q[0], q1 = q[1], q2 = q[2], q3 = q[3];
    float qm = fmaxf(fmaxf(q0, q1), fmaxf(q2, q3));
    float e0 = __expf(q0 - qm), e1 = __expf(q1 - qm);
    float e2 = __expf(q2 - qm), e3 = __expf(q3 - qm);
    float inv = 1.f / (e0 + e1 + e2 + e3);
    float a0 = e0 * inv, a1 = e1 * inv, a2 = e2 * inv, a3 = e3 * inv;

    for (int i = tid; i < 20 * 64; i += 128) {
        int r = i >> 6, c = i & 63;
        sX[r][c] = (r < 19) ? Xin[((size_t)bs * 19 + r) * 64 + c] : 0.f;
    }
    for (int i = tid; i < 64 * 64; i += 128) sW[i >> 6][i & 63] = W[i];

    const size_t pbase   = (size_t)b * 4 * 15 * 361 + (size_t)s * 361;
    const size_t kstride = (size_t)15 * 361;
    for (int i = tid; i < 32 * 20; i += 128) {
        int r = i / 20, c = i % 20;
        float v = 0.f;
        if (r < 19 && c < 19) {
            size_t o = pbase + (size_t)r * 19 + c;
            v = a0 * plv[o] + a1 * plv[o + kstride]
              + a2 * plv[o + 2 * kstride] + a3 * plv[o + 3 * kstride];
        }
        sA[r][c] = v;
    }
    __syncthreads();

    const int lane = tid & 31;
    const int wv   = tid >> 5;
    const int m0   = lane & 15;
    const int h    = lane >> 4;
    const int col  = (wv << 4) + m0;

    // ---- stage 1: AX = Abar(32x20) @ X(20x64)
    v8f acc0 = {}; v8f acc1 = {};
    #pragma unroll
    for (int kk = 0; kk < 20; kk += 4) {
        v2f bf; bf.x = sX[kk + 2 * h + 0][col];
                bf.y = sX[kk + 2 * h + 1][col];
        v2f af0, af1;
        af0.x = sA[m0][kk + 2 * h];      af0.y = sA[m0][kk + 2 * h + 1];
        af1.x = sA[m0 + 16][kk + 2 * h]; af1.y = sA[m0 + 16][kk + 2 * h + 1];
        acc0 = __builtin_amdgcn_wmma_f32_16x16x4_f32(false, af0, false, bf, (short)0, acc0, false, false);
        acc1 = __builtin_amdgcn_wmma_f32_16x16x4_f32(false, af1, false, bf, (short)0, acc1, false, false);
    }
    #pragma unroll
    for (int v = 0; v < 8; ++v) {
        sAX[v + 8 * h][col]      = acc0[v];
        sAX[16 + v + 8 * h][col] = acc1[v];
    }
    __syncthreads();

    // ---- stage 2: G = AX(32x64) @ W(64x64)
    v8f g0 = {}; v8f g1 = {};
    #pragma unroll
    for (int kk = 0; kk < 64; kk += 4) {
        v2f bf; bf.x = sW[kk + 2 * h + 0][col];
                bf.y = sW[kk + 2 * h + 1][col];
        v2f af0, af1;
        af0.x = sAX[m0][kk + 2 * h];      af0.y = sAX[m0][kk + 2 * h + 1];
        af1.x = sAX[m0 + 16][kk + 2 * h]; af1.y = sAX[m0 + 16][kk + 2 * h + 1];
        g0 = __builtin_amdgcn_wmma_f32_16x16x4_f32(false, af0, false, bf, (short)0, g0, false, false);
        g1 = __builtin_amdgcn_wmma_f32_16x16x4_f32(false, af1, false, bf, (short)0, g1, false, false);
    }
    #pragma unroll
    for (int v = 0; v < 8; ++v) {
        int m = v + 8 * h;
        if (m < 19)  Gout[((size_t)bs * 19 + m) * 64 + col] = g0[v];
        int m2 = 16 + v + 8 * h;
        if (m2 < 19) Gout[((size_t)bs * 19 + m2) * 64 + col] = g1[v];
    }
}

// =====================================================================
// Kernel 3: BN statistics per (s,f) over (B=128, N=19) = 2432 samples
// =====================================================================
__global__ void bn_stats_kernel(const float* __restrict__ G, float* __restrict__ stats)
{
    __shared__ float ssum[256], ssq[256];
    int s = blockIdx.x >> 6, f = blockIdx.x & 63;
    float sum = 0.f, sq = 0.f;
    for (int i = threadIdx.x; i < 128 * 19; i += 256) {
        int b = i / 19, n = i % 19;
        float v = G[(((size_t)b * 15 + s) * 19 + n) * 64 + f];
        sum += v; sq += v * v;
    }
    ssum[threadIdx.x] = sum; ssq[threadIdx.x] = sq;
    __syncthreads();
    for (int off = 128; off > 0; off >>= 1) {
        if (threadIdx.x < off) {
            ssum[threadIdx.x] += ssum[threadIdx.x + off];
            ssq[threadIdx.x]  += ssq[threadIdx.x + off];
        }
        __syncthreads();
    }
    if (threadIdx.x == 0) {
        float mean = ssum[0] * (1.f / 2432.f);
        float var  = ssq[0] * (1.f / 2432.f) - mean * mean;  // biased, as reference
        stats[blockIdx.x * 2 + 0] = mean;
        stats[blockIdx.x * 2 + 1] = rsqrtf(var + BN_EPS);
    }
}

// =====================================================================
// Kernel 4: normalize + affine + ReLU (elementwise, in place OK)
// =====================================================================
__global__ void bn_apply_kernel(const float* __restrict__ G, const float* __restrict__ stats,
                                const float* __restrict__ gamma, const float* __restrict__ beta,
                                float* __restrict__ H, int total)
{
    int idx = blockIdx.x * 256 + threadIdx.x;
    if (idx >= total) return;
    int f   = idx & 63;
    int row = idx >> 6;
    int s   = (row / 19) % 15;
    float mean = stats[(s * 64 + f) * 2 + 0];
    float rstd = stats[(s * 64 + f) * 2 + 1];
    float v = (G[idx] - mean) * rstd * gamma[f] + beta[f];
    H[idx] = fmaxf(v, 0.f);
}

// =====================================================================
// Kernel 5: node-mean readout -> out[b,s,f]
// =====================================================================
__global__ void readout_kernel(const float* __restrict__ H, float* __restrict__ out, int total)
{
    int idx = blockIdx.x * 256 + threadIdx.x;
    if (idx >= total) return;
    int f  = idx & 63;
    int bs = idx >> 6;
    float sum = 0.f;
    #pragma unroll
    for (int n = 0; n < 19; ++n) sum += H[((size_t)bs * 19 + n) * 64 + f];
    out[idx] = sum * (1.f / 19.f);
}

extern "C" void kernel_launch(void* const* d_in, const int* in_sizes, int n_in,
                              void* d_out, int out_size, void* d_ws, size_t ws_size,
                              hipStream_t stream) {
    const float* eeg    = (const float*)d_in[0];   // [128,15,19,400]
    const float* plv    = (const float*)d_in[1];   // [128,4,15,19,19]
    const float* c1w    = (const float*)d_in[2];
    const float* c1b    = (const float*)d_in[3];
    const float* c2w    = (const float*)d_in[4];
    const float* c2b    = (const float*)d_in[5];
    const float* W1     = (const float*)d_in[6];
    const float* q1     = (const float*)d_in[7];
    const float* gamma1 = (const float*)d_in[8];
    const float* beta1  = (const float*)d_in[9];
    const float* W2     = (const float*)d_in[10];
    const float* q2     = (const float*)d_in[11];
    const float* gamma2 = (const float*)d_in[12];
    const float* beta2  = (const float*)d_in[13];
    float* out = (float*)d_out;

    const int nSeq  = 128 * 15 * 19;               // 36480
    const int featN = nSeq * 64;                   // 2,334,720 floats
    float* X0    = (float*)d_ws;                   // features / layer-2 output
    float* G     = X0 + featN;                     // layer-1 pre-BN / layer-2 input
    float* stats = G + featN;                      // 960*2 floats
    float* w2T   = stats + 1920;                   // 288*64 floats

    // 0) transpose conv2 weights once
    transpose_w2_kernel<<<dim3(72), dim3(256), 0, stream>>>(c2w, w2T);

    // 1) temporal encoder (all-WMMA, async B2 staging) -> X0
    {
        dim3 grid(nSeq / 4), block(256);
        size_t shmem = 47744 * sizeof(float);      // ~191 KB (CDNA5: 320KB/WG)
        encoder_kernel<<<grid, block, shmem, stream>>>(eeg, c1w, c1b, w2T, c2b, X0, nSeq);
    }

    const int nBS   = 128 * 15;
    const int total = featN;

    // 2) layer 1
    graph_gemm_kernel<<<dim3(nBS), dim3(128), 0, stream>>>(X0, plv, q1, W1, G);
    bn_stats_kernel <<<dim3(15 * 64), dim3(256), 0, stream>>>(G, stats);
    bn_apply_kernel <<<dim3((total + 255) / 256), dim3(256), 0, stream>>>(G, stats, gamma1, beta1, G, total);

    // 3) layer 2
    graph_gemm_kernel<<<dim3(nBS), dim3(128), 0, stream>>>(G, plv, q2, W2, X0);
    bn_stats_kernel <<<dim3(15 * 64), dim3(256), 0, stream>>>(X0, stats);
    bn_apply_kernel <<<dim3((total + 255) / 256), dim3(256), 0, stream>>>(X0, stats, gamma2, beta2, X0, total);

    // 4) readout -> [128,15,64]
    const int outN = 128 * 15 * 64;
    readout_kernel<<<dim3((outN + 255) / 256), dim3(256), 0, stream>>>(X0, out, outN);
}